// DeformableTransformer_90383291777054
// MI455X (gfx1250) — compile-verified
//
#include <hip/hip_runtime.h>
#include <hip/hip_bf16.h>

typedef __attribute__((ext_vector_type(2))) float v2f;
typedef __attribute__((ext_vector_type(8))) float v8f;

#define LEN_IN 13294
#define BATCH  2
#define MROWS  (BATCH * LEN_IN)   // 26588
#define NHEADS 8
#define KDIM   256

// ---------------------------------------------------------------------------
// WMMA GEMM: C[M,N] = A[M,256] @ B[256,N] + bias[N]   (fp32, via
// V_WMMA_F32_16X16X4_F32). Block = 128 threads = 4 waves. Each wave computes a
// 32(M) x 16(N) tile: two 16x16 WMMA accumulators sharing one B fragment, so
// each global B load feeds two matrix ops. Block covers 32(M) x 64(N); the
// 32x256 A tile is staged in LDS (32 KB) and read as ds_load_2addr_b64
// fragments. Only the final M-tile takes the guarded store path.
// ---------------------------------------------------------------------------
__global__ __launch_bounds__(128)
void wmma_gemm_bias_k256(const float* __restrict__ A, const float* __restrict__ Bm,
                         const float* __restrict__ bias, float* __restrict__ C,
                         int M, int N)
{
    __shared__ float As[32 * KDIM];   // 32 KB
    const int tid  = threadIdx.x;
    const int wave = tid >> 5;
    const int lane = tid & 31;
    const int m0   = blockIdx.y * 32;

    // Cooperative load of A tile (32 rows x 256 cols), row-clamped at M-1.
    for (int i = tid; i < 32 * (KDIM / 4); i += 128) {
        int r  = i >> 6;              // 64 float4 per row
        int c4 = (i & 63) << 2;
        int gr = m0 + r;
        if (gr > M - 1) gr = M - 1;
        *(float4*)(&As[r * KDIM + c4]) = *(const float4*)(A + (size_t)gr * KDIM + c4);
    }

    const int n0  = blockIdx.x * 64 + wave * 16;
    const int l15 = lane & 15;
    const int kh  = (lane >> 4) << 1;          // 0 for lanes 0-15, 2 for 16-31
    const float* Bcol = Bm + (size_t)kh * N + n0 + l15;

    // Touch the wave's B column strip (L2-resident): global_prefetch_b8 path.
    __builtin_prefetch((const void*)Bcol, 0, 1);

    __syncthreads();

    const float* Arow0 = &As[(l15     ) * KDIM + kh];
    const float* Arow1 = &As[(l15 + 16) * KDIM + kh];

    v8f acc0 = {0.f, 0.f, 0.f, 0.f, 0.f, 0.f, 0.f, 0.f};
    v8f acc1 = {0.f, 0.f, 0.f, 0.f, 0.f, 0.f, 0.f, 0.f};
#pragma unroll 8
    for (int k0 = 0; k0 < KDIM; k0 += 4) {
        v2f a0, a1, b;
        // A 16x4 fragment: lanes 0-15 -> K=k0,k0+1 ; lanes 16-31 -> K=k0+2,k0+3
        a0[0] = Arow0[k0];  a0[1] = Arow0[k0 + 1];
        a1[0] = Arow1[k0];  a1[1] = Arow1[k0 + 1];
        // B 4x16 fragment: VGPR0 -> row k0(+2 hi lanes), VGPR1 -> row k0+1(+2)
        const float* bp = Bcol + (size_t)k0 * N;
        b[0] = bp[0];
        b[1] = bp[N];
        acc0 = __builtin_amdgcn_wmma_f32_16x16x4_f32(false, a0, false, b,
                                                     (short)0, acc0, false, false);
        acc1 = __builtin_amdgcn_wmma_f32_16x16x4_f32(false, a1, false, b,
                                                     (short)0, acc1, false, false);
    }

    const int   col   = n0 + l15;
    const float bb    = bias[col];
    const int   rbase = m0 + ((lane >> 4) << 3);  // C/D: VGPR j -> M=j (+8 hi lanes)
    float* Cp0 = C + (size_t)rbase * N + col;
    float* Cp1 = Cp0 + (size_t)16 * N;

    if (m0 + 32 <= M) {               // uniform fast path: full tile, no guards
#pragma unroll
        for (int j = 0; j < 8; ++j) {
            Cp0[(size_t)j * N] = acc0[j] + bb;
            Cp1[(size_t)j * N] = acc1[j] + bb;
        }
    } else {                          // only the final M-tile lands here
#pragma unroll
        for (int j = 0; j < 8; ++j) {
            if (rbase + j      < M) Cp0[(size_t)j * N] = acc0[j] + bb;
            if (rbase + 16 + j < M) Cp1[(size_t)j * N] = acc1[j] + bb;
        }
    }
}

// ---------------------------------------------------------------------------
// Softmax over the 16 (level,point) logits of each (row, head) group, in place.
// ---------------------------------------------------------------------------
__global__ __launch_bounds__(256)
void softmax16(float* __restrict__ logits, int rows)
{
    int t = blockIdx.x * 256 + threadIdx.x;           // one thread per (row, head)
    if (t >= rows * NHEADS) return;
    float* p = logits + (size_t)(t >> 3) * 128 + (t & 7) * 16;
    float m = -1e30f;
#pragma unroll
    for (int i = 0; i < 16; ++i) m = fmaxf(m, p[i]);
    float e[16], s = 0.f;
#pragma unroll
    for (int i = 0; i < 16; ++i) { e[i] = __expf(p[i] - m); s += e[i]; }
    float inv = 1.f / s;
#pragma unroll
    for (int i = 0; i < 16; ++i) p[i] = e[i] * inv;
}

// ---------------------------------------------------------------------------
// Bilinear sampling + attention-weighted sum. One wave per (b, query, head);
// the 32 lanes cover the 32 head dims so every gather is a coalesced 128B read
// that stays resident in the 192MB L2 (value tensor is 27MB).
// ---------------------------------------------------------------------------
__device__ __forceinline__ float corner(const float* vb, int xi, int yi, int W, int H)
{
    if (xi < 0 || yi < 0 || xi >= W || yi >= H) return 0.f;
    return vb[(size_t)(yi * W + xi) * 256];
}

__global__ __launch_bounds__(256)
void msdeform_sample(const float* __restrict__ value, const float* __restrict__ offs,
                     const float* __restrict__ aw,    const float* __restrict__ refp,
                     float* __restrict__ outh)
{
    const int HWs[4][2] = {{100,100},{50,50},{25,25},{13,13}};
    const int ST[4]     = {0, 10000, 12500, 13125};

    int gw   = (blockIdx.x * 256 + threadIdx.x) >> 5;   // global wave id
    int lane = threadIdx.x & 31;
    if (gw >= MROWS * NHEADS) return;
    int h   = gw & 7;
    int row = gw >> 3;            // b*Lq + lq
    int b   = row / LEN_IN;

    const float* offrow = offs + (size_t)row * 256 + h * 32;   // (h, lvl, p, xy)
    const float* awrow  = aw   + (size_t)row * 128 + h * 16;   // (h, lvl*p)
    const float* refrow = refp + (size_t)row * 8;              // (lvl, xy)
    const float* vbat   = value + (size_t)b * LEN_IN * 256 + h * 32 + lane;

    float acc = 0.f;
#pragma unroll
    for (int lvl = 0; lvl < 4; ++lvl) {
        const int   H  = HWs[lvl][0], W = HWs[lvl][1];
        const float fh = (float)H,   fw = (float)W;
        const float rx = refrow[lvl * 2 + 0];
        const float ry = refrow[lvl * 2 + 1];
        const float* vb = vbat + (size_t)ST[lvl] * 256;
#pragma unroll
        for (int p = 0; p < 4; ++p) {
            float ox = offrow[(lvl * 4 + p) * 2 + 0];
            float oy = offrow[(lvl * 4 + p) * 2 + 1];
            // loc = ref + off/normalizer; x = loc_x*W - 0.5 == rx*W + ox - 0.5
            float x  = fmaf(rx, fw, ox) - 0.5f;
            float y  = fmaf(ry, fh, oy) - 0.5f;
            float xf = floorf(x), yf = floorf(y);
            float wx = x - xf,    wy = y - yf;
            int   x0 = (int)xf,   y0 = (int)yf;
            float a  = awrow[lvl * 4 + p];

            float v00 = corner(vb, x0,     y0,     W, H);
            float v01 = corner(vb, x0 + 1, y0,     W, H);
            float v10 = corner(vb, x0,     y0 + 1, W, H);
            float v11 = corner(vb, x0 + 1, y0 + 1, W, H);
            float top = v00 + wx * (v01 - v00);
            float bot = v10 + wx * (v11 - v10);
            acc = fmaf(a, top + wy * (bot - top), acc);
        }
    }
    outh[(size_t)row * 256 + h * 32 + lane] = acc;
}

// ---------------------------------------------------------------------------
extern "C" void kernel_launch(void* const* d_in, const int* in_sizes, int n_in,
                              void* d_out, int out_size, void* d_ws, size_t ws_size,
                              hipStream_t stream)
{
    (void)in_sizes; (void)n_in; (void)out_size; (void)ws_size;

    const float* query  = (const float*)d_in[0];
    const float* refp   = (const float*)d_in[1];
    const float* inflat = (const float*)d_in[2];
    // d_in[3] spatial_shapes, d_in[4] level_start_index: compile-time constants
    const float* Wv = (const float*)d_in[5];
    const float* bv = (const float*)d_in[6];
    const float* Ws = (const float*)d_in[7];
    const float* bs = (const float*)d_in[8];
    const float* Wa = (const float*)d_in[9];
    const float* ba = (const float*)d_in[10];
    const float* Wo = (const float*)d_in[11];
    const float* bo = (const float*)d_in[12];
    float* out = (float*)d_out;

    float* ws    = (float*)d_ws;
    float* value = ws;                               // [M, 256]
    float* offs  = value + (size_t)MROWS * 256;      // [M, 256]
    float* awb   = offs  + (size_t)MROWS * 256;      // [M, 128]
    float* outh  = awb   + (size_t)MROWS * 128;      // [M, 256]

    const int mtiles32 = (MROWS + 31) / 32;          // 831
    dim3 blk(128);

    // 1) value = input_flatten @ Wv + bv
    wmma_gemm_bias_k256<<<dim3(256 / 64, mtiles32), blk, 0, stream>>>(inflat, Wv, bv, value, MROWS, 256);
    // 2) sampling offsets = query @ Ws + bs
    wmma_gemm_bias_k256<<<dim3(256 / 64, mtiles32), blk, 0, stream>>>(query, Ws, bs, offs, MROWS, 256);
    // 3) attention logits = query @ Wa + ba
    wmma_gemm_bias_k256<<<dim3(128 / 64, mtiles32), blk, 0, stream>>>(query, Wa, ba, awb, MROWS, 128);
    // 4) softmax over the 16 (level,point) logits per head
    int groups = MROWS * NHEADS;
    softmax16<<<(groups + 255) / 256, 256, 0, stream>>>(awb, MROWS);
    // 5) bilinear sampling + weighted accumulation (one wave per group)
    msdeform_sample<<<(groups * 32 + 255) / 256, 256, 0, stream>>>(value, offs, awb, refp, outh);
    // 6) output projection into d_out
    wmma_gemm_bias_k256<<<dim3(256 / 64, mtiles32), blk, 0, stream>>>(outh, Wo, bo, out, MROWS, 256);
}